// TopKRouter_32478542692666
// MI455X (gfx1250) — compile-verified
//
#include <hip/hip_runtime.h>
#include <cmath>

// ---------------------------------------------------------------------------
// MoE top-2 router for gfx1250 (MI455X).
//   GEMM via V_WMMA_F32_16X16X4_F32, x-tile staging via async global->LDS DMA
//   (ASYNCcnt path) with double buffering.
//   x: [B*N, C] f32, W: [E, C] f32, b: [E] f32
//   outputs: dispatch_mask [tokens, E] f32 ++ combine_weights [tokens, E] f32
// ---------------------------------------------------------------------------

#define NEXP 8
#define TOPK 2
#define TPB 128          // tokens per block in kernel 1
#define STAGE_COLS 64    // K columns staged in LDS per iteration
#define XS_STRIDE 68     // padded row stride (floats): 272B -> 16B aligned, bank-spread

typedef __attribute__((ext_vector_type(2))) float v2f;
typedef __attribute__((ext_vector_type(8))) float v8f;
typedef __attribute__((ext_vector_type(4))) int   v4i;

#if defined(__has_builtin)
#if __has_builtin(__builtin_amdgcn_global_load_async_to_lds_b128) && \
    __has_builtin(__builtin_amdgcn_s_wait_asynccnt)
#define USE_ASYNC_LDS 1
#endif
#endif

#ifdef USE_ASYNC_LDS
typedef __attribute__((address_space(1))) v4i gv4i;  // global int4
typedef __attribute__((address_space(3))) v4i lv4i;  // LDS int4
#endif

// ---------------------------------------------------------------------------
// Kernel 1: router logits via WMMA, softmax-free top-2 weights, in-order
// per-(rank,expert) local ranks + per-block histograms.
// 256 threads = 8 waves; each wave owns 16 tokens -> 128 tokens / block.
// ---------------------------------------------------------------------------
__global__ __launch_bounds__(256) void router_wmma_top2(
    const float* __restrict__ x, const float* __restrict__ W,
    const float* __restrict__ bias, int C,
    unsigned* __restrict__ tokIdx,   // packed e0 | e1<<8
    float2*   __restrict__ tokW,     // renormalized (w0, w1)
    unsigned* __restrict__ tokRank,  // packed local ranks r0 | r1<<16
    int*      __restrict__ blkCnt)   // [numBlocks][16] counts per (k*8+e)
{
  __shared__ alignas(16) float xs[2][TPB][XS_STRIDE];          // 2x34816 B x-tiles
  __shared__ alignas(16) float wt2[2][STAGE_COLS / 2][16][2];  // 2x4096 B W^T K-pairs
  __shared__ alignas(16) float lg[8][16][16];                  // 8192 B per-wave logits
  __shared__ int cnt4[4][16];                                  // per-wave (k,e) counts

  const int tid = threadIdx.x;
  const int wave = tid >> 5;
  const int lane = tid & 31;
  const int half = lane >> 4;   // 0: K=0,1 ; 1: K=2,3 (A/B fragment lane-halves)
  const int l16  = lane & 15;
  const int tokenBase = blockIdx.x * TPB;
  const int waveRow = wave * 16;

  // ---- stagers -------------------------------------------------------------
  // x tile: rows 0..127, cols [stage, stage+64). Async DMA global -> LDS.
  auto issue_xs = [&](int stage, int buf) {
    const int r0 = tid >> 4;          // 0..15
    const int c4 = (tid & 15) << 2;   // 0..60 step 4
    #pragma unroll
    for (int rr = 0; rr < TPB; rr += 16) {
      const float* gp = x + (size_t)(tokenBase + r0 + rr) * C + stage + c4;
      float* lp = &xs[buf][r0 + rr][c4];
#ifdef USE_ASYNC_LDS
      __builtin_amdgcn_global_load_async_to_lds_b128(
          (gv4i*)(size_t)gp, (lv4i*)(unsigned)(size_t)lp, 0, 0);
#else
      *(float4*)lp = *(const float4*)gp;
#endif
    }
  };
  // W^T K-pairs: wt2[buf][cp][e] = {W[e][stage+2cp], W[e][stage+2cp+1]}, e>=8 -> 0
  auto issue_wt2 = [&](int stage, int buf) {
    for (int idx = tid; idx < (STAGE_COLS / 2) * 16; idx += 256) {
      const int e  = idx & 15;
      const int cp = idx >> 4;
      float vx = 0.f, vy = 0.f;
      if (e < NEXP) {
        const float* wp = W + (size_t)e * C + stage + 2 * cp;
        vx = wp[0];
        vy = wp[1];
      }
      wt2[buf][cp][e][0] = vx;
      wt2[buf][cp][e][1] = vy;
    }
  };

  v8f acc = {0.f, 0.f, 0.f, 0.f, 0.f, 0.f, 0.f, 0.f};

  const int NS = C / STAGE_COLS;
  issue_wt2(0, 0);
  issue_xs(0, 0);

  for (int s = 0; s < NS; ++s) {
    const int buf = s & 1;
#ifdef USE_ASYNC_LDS
    __builtin_amdgcn_s_wait_asynccnt(0);   // this wave's DMA for stage s done
#endif
    __syncthreads();                       // all waves' tiles for stage s visible

    if (s + 1 < NS) {                      // prefetch stage s+1 into other buffer
      issue_wt2((s + 1) * STAGE_COLS, buf ^ 1);
      issue_xs((s + 1) * STAGE_COLS, buf ^ 1);
    }

    // ---- 16 x V_WMMA_F32_16X16X4_F32 over this 64-column stage ----
    #pragma unroll
    for (int i = 0; i < STAGE_COLS / 4; ++i) {
      // A 16x4: lanes 0-15 = rows, K = 4i+0..1 ; lanes 16-31: K = 4i+2..3
      const v2f a = *(const v2f*)&xs[buf][waveRow + l16][4 * i + half * 2];
      // B 4x16: lanes = expert column; lane-half selects K-pair (cp = 2i + half)
      const v2f b = *(const v2f*)&wt2[buf][2 * i + half][l16][0];
      acc = __builtin_amdgcn_wmma_f32_16x16x4_f32(
          false, a, false, b, (short)0, acc, false, false);
    }
  }

  // ---- spill D (16x16 f32 layout: VGPR r -> M = half*8 + r, N = l16) ----
  __syncthreads();
  #pragma unroll
  for (int r = 0; r < 8; ++r)
    lg[wave][half * 8 + r][l16] = acc[r];
  __syncthreads();

  // ---- per-token top-2 + renormalized weights (threads 0..127, token = tid) ----
  int e0 = 0, e1 = 0;
  float w0 = 0.f, w1 = 0.f;
  if (tid < TPB) {
    const int wv = tid >> 4, m = tid & 15;
    float p[NEXP];
    #pragma unroll
    for (int e = 0; e < NEXP; ++e) p[e] = lg[wv][m][e] + bias[e];
    float b0 = p[0];
    #pragma unroll
    for (int e = 1; e < NEXP; ++e)
      if (p[e] > b0) { b0 = p[e]; e0 = e; }   // lowest index wins ties (== lax.top_k)
    float b1 = -INFINITY; e1 = (e0 == 0) ? 1 : 0;
    #pragma unroll
    for (int e = 0; e < NEXP; ++e)
      if (e != e0 && p[e] > b1) { b1 = p[e]; e1 = e; }
    // softmax normalizer cancels in topk_w / sum(topk_w):
    const float t = expf(b1 - b0);            // <= 1, numerically safe
    w0 = 1.f / (1.f + t);
    w1 = 1.f - w0;
  }

  // ---- in-order local ranks per (rank k, expert e); token order == tid ----
  unsigned r0 = 0, r1 = 0;
  if (tid < TPB) {  // waves 0..3 fully active -> ballots see full wave32
    const unsigned long long lt = (1ull << lane) - 1ull;
    #pragma unroll
    for (int k = 0; k < TOPK; ++k) {
      const int my_e = (k == 0) ? e0 : e1;
      #pragma unroll
      for (int e = 0; e < NEXP; ++e) {
        const unsigned long long bal = __ballot(my_e == e);
        if (lane == 0) cnt4[wave][k * 8 + e] = __popcll(bal);
        if (my_e == e) {
          const unsigned rk = (unsigned)__popcll(bal & lt);
          if (k == 0) r0 = rk; else r1 = rk;
        }
      }
    }
  }
  __syncthreads();

  if (tid < TPB) {
    #pragma unroll
    for (int w = 0; w < 4; ++w) {
      if (w < wave) {
        r0 += (unsigned)cnt4[w][e0];
        r1 += (unsigned)cnt4[w][8 + e1];
      }
    }
    tokIdx[tokenBase + tid]  = (unsigned)e0 | ((unsigned)e1 << 8);
    tokW[tokenBase + tid]    = make_float2(w0, w1);
    tokRank[tokenBase + tid] = r0 | (r1 << 16);
  }
  if (tid < 16) {
    blkCnt[blockIdx.x * 16 + tid] =
        cnt4[0][tid] + cnt4[1][tid] + cnt4[2][tid] + cnt4[3][tid];
  }
}

// ---------------------------------------------------------------------------
// Kernel 2: single-block exclusive scan of per-block histograms along blocks,
// independently for each of the 16 (rank, expert) streams. numBlocks <= 256.
// ---------------------------------------------------------------------------
__global__ __launch_bounds__(256) void scan_block_counts(
    const int* __restrict__ blkCnt, int* __restrict__ blkOff, int numBlocks)
{
  __shared__ int sc[256][17];
  const int t = threadIdx.x;
  int v[16], orig[16];
  #pragma unroll
  for (int p = 0; p < 16; ++p) {
    v[p] = (t < numBlocks) ? blkCnt[t * 16 + p] : 0;
    orig[p] = v[p];
  }
  for (int d = 1; d < 256; d <<= 1) {
    #pragma unroll
    for (int p = 0; p < 16; ++p) sc[t][p] = v[p];
    __syncthreads();
    if (t >= d) {
      #pragma unroll
      for (int p = 0; p < 16; ++p) v[p] += sc[t - d][p];
    }
    __syncthreads();
  }
  if (t < numBlocks) {
    #pragma unroll
    for (int p = 0; p < 16; ++p) blkOff[t * 16 + p] = v[p] - orig[p];  // exclusive
  }
}

// ---------------------------------------------------------------------------
// Kernel 3: capacity check + emit dispatch_mask and combine_weights.
// ---------------------------------------------------------------------------
__global__ __launch_bounds__(256) void emit_router_outputs(
    const unsigned* __restrict__ tokIdx, const float2* __restrict__ tokW,
    const unsigned* __restrict__ tokRank, const int* __restrict__ blkOff,
    float* __restrict__ outDisp, float* __restrict__ outComb,
    int tokens, int capacity)
{
  const int t = blockIdx.x * 256 + threadIdx.x;
  if (t >= tokens) return;
  const unsigned pk = tokIdx[t];
  const int e0 = pk & 0xff, e1 = (pk >> 8) & 0xff;
  const float2 w = tokW[t];
  const unsigned rk = tokRank[t];
  const int sb = t / TPB;
  const int p0 = blkOff[sb * 16 + e0] + (int)(rk & 0xffffu);
  const int p1 = blkOff[sb * 16 + 8 + e1] + (int)(rk >> 16);
  const bool a0 = p0 < capacity;
  const bool a1 = p1 < capacity;

  float disp[NEXP] = {0.f, 0.f, 0.f, 0.f, 0.f, 0.f, 0.f, 0.f};
  float comb[NEXP] = {0.f, 0.f, 0.f, 0.f, 0.f, 0.f, 0.f, 0.f};
  if (a0) { disp[e0] = 1.f; comb[e0] = w.x; }
  if (a1) { disp[e1] = 1.f; comb[e1] = w.y; }
  const float s = (a0 ? w.x : 0.f) + (a1 ? w.y : 0.f);
  const float inv = 1.f / fmaxf(s, 1e-6f);
  #pragma unroll
  for (int e = 0; e < NEXP; ++e) comb[e] *= inv;

  float* dp = outDisp + (size_t)t * NEXP;
  float* cp = outComb + (size_t)t * NEXP;
  *(float4*)(dp)     = make_float4(disp[0], disp[1], disp[2], disp[3]);
  *(float4*)(dp + 4) = make_float4(disp[4], disp[5], disp[6], disp[7]);
  *(float4*)(cp)     = make_float4(comb[0], comb[1], comb[2], comb[3]);
  *(float4*)(cp + 4) = make_float4(comb[4], comb[5], comb[6], comb[7]);
}

// ---------------------------------------------------------------------------
extern "C" void kernel_launch(void* const* d_in, const int* in_sizes, int n_in,
                              void* d_out, int out_size, void* d_ws, size_t ws_size,
                              hipStream_t stream) {
  const float* x = (const float*)d_in[0];
  const float* W = (const float*)d_in[1];
  const float* b = (const float*)d_in[2];

  const int C = in_sizes[1] / NEXP;          // 1024
  const int tokens = in_sizes[0] / C;        // 32768
  const int numBlocks = tokens / TPB;        // 256
  const int capacity =
      (int)std::ceil(1.25 * (double)tokens * (double)TOPK / (double)NEXP);  // 10240

  float* outDisp = (float*)d_out;
  float* outComb = outDisp + (size_t)tokens * NEXP;

  // workspace layout
  unsigned* tokIdx  = (unsigned*)d_ws;                    // tokens u32
  float2*   tokW    = (float2*)(tokIdx + tokens);         // tokens float2
  unsigned* tokRank = (unsigned*)(tokW + tokens);         // tokens u32
  int*      blkCnt  = (int*)(tokRank + tokens);           // numBlocks*16
  int*      blkOff  = blkCnt + numBlocks * 16;            // numBlocks*16

  router_wmma_top2<<<numBlocks, 256, 0, stream>>>(x, W, b, C, tokIdx, tokW,
                                                  tokRank, blkCnt);
  scan_block_counts<<<1, 256, 0, stream>>>(blkCnt, blkOff, numBlocks);
  emit_router_outputs<<<(tokens + 255) / 256, 256, 0, stream>>>(
      tokIdx, tokW, tokRank, blkOff, outDisp, outComb, tokens, capacity);
}